// SimpleQuantizer_76364518523192
// MI455X (gfx1250) — compile-verified
//
#include <hip/hip_runtime.h>
#include <hip/hip_bf16.h>
#include <math.h>
#include <stdint.h>

typedef __attribute__((ext_vector_type(2))) float v2f;
typedef __attribute__((ext_vector_type(8))) float v8f;

#define BM 128
#define BN 128
#define BK 16
#define APAD 4  // A row stride 20 floats = 80B: 16B-aligned, conflict-free b64 frag reads

// Low 32 bits of a flat shared-aperture address == LDS byte offset (ISA §10.2).
__device__ __forceinline__ uint32_t lds_off(const void* p) {
    return (uint32_t)(uintptr_t)p;
}

// ---------------------------------------------------------------------------
// Pass 1: w_rot = W (MxK) @ R (KxN), f32, via V_WMMA_F32_16X16X4_F32.
// Block tile 128x128, K-tile 16, 8 waves, each wave owns a 32x64 subtile
// (2x4 WMMA accumulators). LDS is double-buffered and staged with
// GLOBAL_LOAD_ASYNC_TO_LDS_B128 (ASYNCcnt), overlapping HBM/L2 traffic with
// the matrix pipe.
// ---------------------------------------------------------------------------
__global__ __launch_bounds__(256) void quant_gemm_f32_wmma(
    const float* __restrict__ W, const float* __restrict__ R,
    float* __restrict__ out, int M, int N, int K)
{
    __shared__ float As[2][BM][BK + APAD];   // 2 x 10 KB
    __shared__ float Bs[2][BK][BN];          // 2 x  8 KB

    const int tid    = threadIdx.x;
    const int wave   = tid >> 5;
    const int lane   = tid & 31;
    const int wave_m = wave >> 1;            // 0..3 -> 32-row chunk
    const int wave_n = wave & 1;             // 0..1 -> 64-col chunk
    const int lrow   = lane & 15;
    const int lhi    = lane >> 4;

    const int block_m = blockIdx.y * BM;
    const int block_n = blockIdx.x * BN;

    // Per-thread staging slots: 2 x b128 for A (128x16) and 2 x b128 for B (16x128).
    const int ar0 = tid >> 2;                // 0..63   (second slot: +64)
    const int ac0 = (tid & 3) * 4;           // K sub-column (floats)
    const int br0 = tid >> 5;                // 0..7    (second slot: +8)
    const int bc0 = (tid & 31) * 4;          // N sub-column (floats)

    const uint64_t Wg = (uint64_t)(uintptr_t)W;
    const uint64_t Rg = (uint64_t)(uintptr_t)R;

    // Issue 4 async b128 global->LDS lane transfers for K-tile k0 into buffer `buf`.
    auto stage_async = [&](int buf, int k0) {
        uint32_t ldsA0 = lds_off(&As[buf][ar0][ac0]);
        uint32_t ldsA1 = lds_off(&As[buf][ar0 + 64][ac0]);
        uint32_t voA0  = (uint32_t)((uint32_t)(block_m + ar0)      * (uint32_t)K + (uint32_t)(k0 + ac0)) * 4u;
        uint32_t voA1  = (uint32_t)((uint32_t)(block_m + ar0 + 64) * (uint32_t)K + (uint32_t)(k0 + ac0)) * 4u;
        asm volatile("global_load_async_to_lds_b128 %0, %1, %2"
                     :: "v"(ldsA0), "v"(voA0), "s"(Wg) : "memory");
        asm volatile("global_load_async_to_lds_b128 %0, %1, %2"
                     :: "v"(ldsA1), "v"(voA1), "s"(Wg) : "memory");

        uint32_t ldsB0 = lds_off(&Bs[buf][br0][bc0]);
        uint32_t ldsB1 = lds_off(&Bs[buf][br0 + 8][bc0]);
        uint32_t voB0  = (uint32_t)((uint32_t)(k0 + br0)     * (uint32_t)N + (uint32_t)(block_n + bc0)) * 4u;
        uint32_t voB1  = (uint32_t)((uint32_t)(k0 + br0 + 8) * (uint32_t)N + (uint32_t)(block_n + bc0)) * 4u;
        asm volatile("global_load_async_to_lds_b128 %0, %1, %2"
                     :: "v"(ldsB0), "v"(voB0), "s"(Rg) : "memory");
        asm volatile("global_load_async_to_lds_b128 %0, %1, %2"
                     :: "v"(ldsB1), "v"(voB1), "s"(Rg) : "memory");
    };

    v8f acc[2][4];
#pragma unroll
    for (int mt = 0; mt < 2; ++mt)
#pragma unroll
        for (int nt = 0; nt < 4; ++nt)
            acc[mt][nt] = (v8f){0.f, 0.f, 0.f, 0.f, 0.f, 0.f, 0.f, 0.f};

    // Prologue: stage tile 0 and wait for it.
    stage_async(0, 0);
    asm volatile("s_wait_asynccnt 0" ::: "memory");
    __syncthreads();

    int buf = 0;
    for (int k0 = 0; k0 < K; k0 += BK) {
        // Kick off the next tile's async copies; they stream while we do WMMA.
        const int kn = k0 + BK;
        if (kn < K) stage_async(buf ^ 1, kn);

#pragma unroll
        for (int kk = 0; kk < BK; kk += 4) {
            const int kfrag = kk + 2 * lhi;
            v2f afrag[2];
#pragma unroll
            for (int mt = 0; mt < 2; ++mt) {
                const int m = wave_m * 32 + mt * 16 + lrow;
                afrag[mt] = *reinterpret_cast<const v2f*>(&As[buf][m][kfrag]);
            }
            v2f bfrag[4];
#pragma unroll
            for (int nt = 0; nt < 4; ++nt) {
                const int n = wave_n * 64 + nt * 16 + lrow;
                bfrag[nt].x = Bs[buf][kfrag][n];
                bfrag[nt].y = Bs[buf][kfrag + 1][n];
            }
#pragma unroll
            for (int mt = 0; mt < 2; ++mt)
#pragma unroll
                for (int nt = 0; nt < 4; ++nt)
                    acc[mt][nt] = __builtin_amdgcn_wmma_f32_16x16x4_f32(
                        false, afrag[mt], false, bfrag[nt],
                        (short)0, acc[mt][nt], false, false);
        }

        // Next tile must be fully in LDS, and every wave must be done reading
        // `buf`, before we flip.
        asm volatile("s_wait_asynccnt 0" ::: "memory");
        __syncthreads();
        buf ^= 1;
    }

    // Store accumulators: D layout M = r + 8*lhi, N = lrow (16 lanes -> 64B rows).
#pragma unroll
    for (int mt = 0; mt < 2; ++mt)
#pragma unroll
        for (int nt = 0; nt < 4; ++nt)
#pragma unroll
            for (int r = 0; r < 8; ++r) {
                const int m = block_m + wave_m * 32 + mt * 16 + r + 8 * lhi;
                const int n = block_n + wave_n * 64 + nt * 16 + lrow;
                out[(size_t)m * N + n] = acc[mt][nt][r];
            }
}

// ---------------------------------------------------------------------------
// Pass 2: per-row min/max (permutation-invariant), then gathered fake-quant.
// One 256-thread workgroup per output row.
// ---------------------------------------------------------------------------
__global__ __launch_bounds__(256) void quant_row_kernel(
    const float* __restrict__ wrot, const int* __restrict__ perm,
    float* __restrict__ out, int N)
{
    const int row = blockIdx.x;
    const float* src = wrot + (size_t)row * N;

    float vmin =  INFINITY;
    float vmax = -INFINITY;
    for (int j = threadIdx.x; j < N; j += 256) {
        const float v = src[j];
        vmin = fminf(vmin, v);
        vmax = fmaxf(vmax, v);
    }
#pragma unroll
    for (int off = 16; off > 0; off >>= 1) {
        vmin = fminf(vmin, __shfl_xor(vmin, off, 32));
        vmax = fmaxf(vmax, __shfl_xor(vmax, off, 32));
    }
    __shared__ float smin[8], smax[8];
    const int wv = threadIdx.x >> 5;
    if ((threadIdx.x & 31) == 0) { smin[wv] = vmin; smax[wv] = vmax; }
    __syncthreads();
    float rmin = smin[0], rmax = smax[0];
#pragma unroll
    for (int w = 1; w < 8; ++w) {
        rmin = fminf(rmin, smin[w]);
        rmax = fmaxf(rmax, smax[w]);
    }

    const float qmax  = 255.0f;
    const float scale = fmaxf((rmax - rmin) / qmax, 1e-8f);
    const float inv_s = 1.0f / scale;
    const float zp    = -rmin * inv_s;

    for (int j = threadIdx.x; j < N; j += 256) {
        const float v = src[perm[j]];
        const float q = fminf(fmaxf(rintf(v * inv_s + zp), 0.0f), qmax);
        out[(size_t)row * N + j] = (q - zp) * scale;
    }
}

extern "C" void kernel_launch(void* const* d_in, const int* in_sizes, int n_in,
                              void* d_out, int out_size, void* d_ws, size_t ws_size,
                              hipStream_t stream) {
    const float* W    = (const float*)d_in[0];
    const float* R    = (const float*)d_in[1];
    const int*   perm = (const int*)d_in[2];
    float*       out  = (float*)d_out;
    float*       ws   = (float*)d_ws;   // holds W@R: M*N floats (128 MB)

    const int K = in_sizes[2];          // 4096 (perm length = in_features)
    const int M = in_sizes[0] / K;      // 8192
    const int N = K;                    // 4096

    dim3 grid1(N / BN, M / BM);         // 32 x 64 workgroups
    quant_gemm_f32_wmma<<<grid1, dim3(256), 0, stream>>>(W, R, ws, M, N, K);
    quant_row_kernel<<<dim3(M), dim3(256), 0, stream>>>(ws, perm, out, N);
}